// ProposalGenerator_57114475102297
// MI455X (gfx1250) — compile-verified
//
#include <hip/hip_runtime.h>
#include <hip/hip_bf16.h>
#include <stdint.h>

// ---------------- problem constants (from reference) ----------------
#define H_      128
#define W_      128
#define A_      9
#define HW_     (H_ * W_)        // 16384
#define N_      (HW_ * A_)       // 147456 proposals
#define NPAD_   262144           // next pow2 >= N_ for bitonic sort
#define PRE_    12000
#define POST_   2000
#define MASKW_  ((PRE_ + 31) / 32)   // 375 mask words per row (12000/32 exactly)
#define NMS_T   0.7f

typedef __attribute__((ext_vector_type(2))) float v2f;
typedef __attribute__((ext_vector_type(8))) float v8f;
typedef __attribute__((ext_vector_type(4))) unsigned int v4u;
typedef __attribute__((ext_vector_type(8))) int v8i;
typedef __attribute__((ext_vector_type(4))) int v4i;

#if __has_builtin(__builtin_amdgcn_tensor_load_to_lds) && \
    __has_builtin(__builtin_amdgcn_s_wait_tensorcnt)
#define USE_TDM 1
#endif

// Base anchors (precomputed from generate_base_anchors(), exact values)
__constant__ float c_bx1[9] = {-84.f,-176.f,-360.f,-56.f,-120.f,-248.f,-36.f,-80.f,-168.f};
__constant__ float c_by1[9] = {-40.f,-88.f,-184.f,-56.f,-120.f,-248.f,-80.f,-168.f,-344.f};
__constant__ float c_bx2[9] = { 99.f, 191.f, 375.f, 71.f, 135.f, 263.f, 51.f, 95.f, 183.f};
__constant__ float c_by2[9] = { 55.f, 103.f, 199.f, 71.f, 135.f, 263.f, 95.f, 183.f, 359.f};

// ---------------- K0: zero the NMS bitmask + keep_count ----------------
__global__ void zero_mask_kernel(unsigned* __restrict__ mask, int n, int* __restrict__ keep_count) {
  int i = blockIdx.x * blockDim.x + threadIdx.x;
  if (i < n) mask[i] = 0u;
  if (i == 0) *keep_count = 0;
}

// ---------------- K1: decode boxes, clip, validity, build sort keys ----------------
__global__ void decode_kernel(const float* __restrict__ deltas,
                              const float* __restrict__ scores,
                              const float* __restrict__ iminfo,
                              float* __restrict__ X1, float* __restrict__ Y1,
                              float* __restrict__ X2, float* __restrict__ Y2,
                              unsigned long long* __restrict__ keys) {
  int i = blockIdx.x * blockDim.x + threadIdx.x;
  if (i >= NPAD_) return;
  if (i >= N_) { keys[i] = 0ull; return; }   // pad keys sort below every real entry

  int a = i % A_;
  int s = i / A_;                 // pixel index h*W + w
  int w = s & (W_ - 1);
  int h = s >> 7;

  float dx = deltas[(4 * a + 0) * HW_ + s];
  float dy = deltas[(4 * a + 1) * HW_ + s];
  float dw = deltas[(4 * a + 2) * HW_ + s];
  float dh = deltas[(4 * a + 3) * HW_ + s];
  float sc = scores[(9 + a) * HW_ + s];

  float sx = (float)w * 16.f;
  float sy = (float)h * 16.f;
  float x1a = sx + c_bx1[a];
  float y1a = sy + c_by1[a];
  float wsa = c_bx2[a] - c_bx1[a] + 1.f;
  float hsa = c_by2[a] - c_by1[a] + 1.f;
  float cx  = x1a + 0.5f * wsa;
  float cy  = y1a + 0.5f * hsa;

  float pcx = dx * wsa + cx;
  float pcy = dy * hsa + cy;
  float pw  = __expf(dw) * wsa;
  float ph  = __expf(dh) * hsa;

  float im_h = iminfo[0], im_w = iminfo[1], im_s = iminfo[2];
  float x1 = fminf(fmaxf(pcx - 0.5f * pw, 0.f), im_w - 1.f);
  float y1 = fminf(fmaxf(pcy - 0.5f * ph, 0.f), im_h - 1.f);
  float x2 = fminf(fmaxf(pcx + 0.5f * pw, 0.f), im_w - 1.f);
  float y2 = fminf(fmaxf(pcy + 0.5f * ph, 0.f), im_h - 1.f);

  float min_size = 16.f * im_s;
  bool valid = (x2 - x1 + 1.f >= min_size) && (y2 - y1 + 1.f >= min_size);
  float ms = valid ? sc : -INFINITY;

  X1[i] = x1; Y1[i] = y1; X2[i] = x2; Y2[i] = y2;

  // monotonic float->uint, descending-stable: low bits = ~i so ties pick smaller i
  unsigned ub = __float_as_uint(ms);
  unsigned su = (ub & 0x80000000u) ? ~ub : (ub | 0x80000000u);
  keys[i] = ((unsigned long long)su << 32) | (unsigned)(~(unsigned)i);
}

// ---------------- K2a: global bitonic step (descending), strides >= 2048 ----------------
__global__ void bitonic_step(unsigned long long* __restrict__ keys, unsigned j, unsigned k) {
  unsigned i = blockIdx.x * blockDim.x + threadIdx.x;
  unsigned ixj = i ^ j;
  if (i < NPAD_ && ixj > i) {
    unsigned long long a = keys[i];
    unsigned long long b = keys[ixj];
    bool up = (i & k) == 0;                 // "up" segments sort descending
    if (up ? (a < b) : (a > b)) { keys[i] = b; keys[ixj] = a; }
  }
}

// ---------------- K2b: LDS-resident bitonic phases (strides <= 1024) ----------------
// kmerge == 0 : full local presort (k = 2..2048)
// kmerge != 0 : merge tail for that k (j = 1024..1)
// 2048 keys * 8B = 16 KB LDS per block; compare-exchange runs on the DS path.
__global__ void __launch_bounds__(1024)
bitonic_local(unsigned long long* __restrict__ keys, unsigned kmerge) {
  __shared__ unsigned long long sk[2048];
  const unsigned base = blockIdx.x * 2048u;
  sk[threadIdx.x]          = keys[base + threadIdx.x];
  sk[threadIdx.x + 1024u]  = keys[base + threadIdx.x + 1024u];
  __syncthreads();

  const unsigned klo = (kmerge == 0u) ? 2u : kmerge;
  const unsigned khi = (kmerge == 0u) ? 2048u : kmerge;
  for (unsigned k = klo; k <= khi; k <<= 1) {
    unsigned jstart = (kmerge == 0u) ? (k >> 1) : 1024u;
    for (unsigned j = jstart; j > 0; j >>= 1) {
#pragma unroll
      for (int e0 = 0; e0 < 2; ++e0) {
        unsigned e = threadIdx.x + (unsigned)e0 * 1024u;
        unsigned ixj = e ^ j;
        if (ixj > e) {
          unsigned long long a = sk[e];
          unsigned long long b = sk[ixj];
          bool up = (((base + e) & k) == 0u);
          if (up ? (a < b) : (a > b)) { sk[e] = b; sk[ixj] = a; }
        }
      }
      __syncthreads();
    }
    if (kmerge != 0u) break;
  }

  keys[base + threadIdx.x]         = sk[threadIdx.x];
  keys[base + threadIdx.x + 1024u] = sk[threadIdx.x + 1024u];
}

// ---------------- K3: gather top-12000 boxes/scores/areas ----------------
// TBOX is contiguous [5][PRE_]: rows = x1,y1,x2,y2,area (exact stride for the TDM D#).
__global__ void gather_top(const unsigned long long* __restrict__ keys,
                           const float* __restrict__ X1, const float* __restrict__ Y1,
                           const float* __restrict__ X2, const float* __restrict__ Y2,
                           float* __restrict__ TBOX, float* __restrict__ TS,
                           int* __restrict__ TVALID) {
  int t = blockIdx.x * blockDim.x + threadIdx.x;
  if (t >= PRE_) return;
  unsigned long long key = keys[t];
  unsigned su  = (unsigned)(key >> 32);
  unsigned idx = ~(unsigned)(key & 0xFFFFFFFFu);
  if (idx >= (unsigned)N_) idx = 0;
  unsigned ub = (su & 0x80000000u) ? (su ^ 0x80000000u) : ~su;  // inverse key map
  float sc = __uint_as_float(ub);
  float x1 = X1[idx], y1 = Y1[idx], x2 = X2[idx], y2 = Y2[idx];
  TBOX[0 * PRE_ + t] = x1;
  TBOX[1 * PRE_ + t] = y1;
  TBOX[2 * PRE_ + t] = x2;
  TBOX[3 * PRE_ + t] = y2;
  TBOX[4 * PRE_ + t] = (x2 - x1 + 1.f) * (y2 - y1 + 1.f);
  TS[t] = sc;
  TVALID[t] = (sc > -INFINITY) ? 1 : 0;
}

// ---------------- K4: NMS suppression bitmask ----------------
// One wave32 owns a 16-row x 32-col tile == one full mask word per row:
//   - 2x V_WMMA_F32_16X16X4_F32 compute the union term area_i + area_j
//     (rank-2 broadcast add on the matrix pipe, co-executing with VALU min/max)
//   - row boxes arrive in LDS via the Tensor Data Mover (2D tile 16x5, stride 12000)
//   - word composed from two wave32 ballots, committed with a plain store (no atomics)
__global__ void __launch_bounds__(256)
nms_mask_kernel(const float* __restrict__ TBOX, unsigned* __restrict__ mask) {
  __shared__ float ROWS[5][16];   // only LDS object in this kernel -> LDS offset 0
  const int r0 = blockIdx.y * 16;

#ifdef USE_TDM
  {
    const float* gsrc = TBOX + r0;                       // row j at gsrc + j*PRE_
    unsigned long long ga = (unsigned long long)(size_t)gsrc;
    v4u g0;
    g0.x = 1u;                                           // count=1 (valid user D#)
    g0.y = 0u;                                           // lds_addr = 0 (ROWS)
    g0.z = (unsigned)ga;                                 // global_addr[31:0]
    g0.w = ((unsigned)(ga >> 32) & 0x01FFFFFFu)          // global_addr[56:32]
           | 0x80000000u;                                // type = 2 ("image")
    v8i g1;
    g1[0] = 0x00020000;                                  // data_size = 4B (code 2)
    g1[1] = (int)(16u << 16);                            // tensor_dim0 = 16
    g1[2] = (int)(5u << 16);                             // tensor_dim1 = 5
    g1[3] = (int)(16u << 16);                            // tile_dim0 = 16
    g1[4] = 5;                                           // tile_dim1 = 5
    g1[5] = PRE_;                                        // tensor_dim0_stride = 12000
    g1[6] = 0;
    g1[7] = 0;
    v4i gz = {0, 0, 0, 0};
    v8i gz8 = {0, 0, 0, 0, 0, 0, 0, 0};
    if (threadIdx.x < 32) {                              // wave 0 issues the DMA
      // amdgpu-toolchain (clang-23) 6-arg form:
      // (v4u g0, v8i g1, v4i g2, v4i g3, v8i extra, i32 cpol)
      __builtin_amdgcn_tensor_load_to_lds(g0, g1, gz, gz, gz8, 0);
      __builtin_amdgcn_s_wait_tensorcnt(0);
    }
  }
#else
  if (threadIdx.x < 80) {
    ((float*)ROWS)[threadIdx.x] =
        TBOX[(threadIdx.x >> 4) * PRE_ + r0 + (threadIdx.x & 15)];
  }
#endif
  __syncthreads();

  const int wave = threadIdx.x >> 5;
  const int lane = threadIdx.x & 31;
  const int c0 = (blockIdx.x * 8 + wave) * 32;    // 32 columns == 1 mask word
  if (c0 >= PRE_) return;                         // wave-uniform exit
  if (c0 + 31 <= r0) return;                      // tile entirely at/below diagonal

  const bool lo = lane < 16;
  const int l15 = lane & 15;
  const float arow = lo ? ROWS[4][l15] : 0.f;

  // A (shared by both halves): A[m][0]=area_row[m], A[m][1]=1; lanes>=16 hold K=2,3 (zeros)
  v2f av;
  av.x = lo ? arow : 0.f;
  av.y = lo ? 1.f : 0.f;
  // B per half: B[0][n]=1, B[1][n]=area_col[n]
  v2f bv0, bv1;
  bv0.x = lo ? 1.f : 0.f;
  bv0.y = lo ? TBOX[4 * PRE_ + c0 + l15] : 0.f;
  bv1.x = bv0.x;
  bv1.y = lo ? TBOX[4 * PRE_ + c0 + 16 + l15] : 0.f;
  v8f cz = {0.f, 0.f, 0.f, 0.f, 0.f, 0.f, 0.f, 0.f};
  v8f asum0 = __builtin_amdgcn_wmma_f32_16x16x4_f32(false, av, false, bv0, (short)0, cz, false, false);
  v8f asum1 = __builtin_amdgcn_wmma_f32_16x16x4_f32(false, av, false, bv1, (short)0, cz, false, false);

  // per-lane column boxes for both halves
  const int ca = c0 + l15;
  const int cb = c0 + 16 + l15;
  const float ax1 = TBOX[0 * PRE_ + ca], ay1 = TBOX[1 * PRE_ + ca];
  const float ax2 = TBOX[2 * PRE_ + ca], ay2 = TBOX[3 * PRE_ + ca];
  const float bx1 = TBOX[0 * PRE_ + cb], by1 = TBOX[1 * PRE_ + cb];
  const float bx2 = TBOX[2 * PRE_ + cb], by2 = TBOX[3 * PRE_ + cb];

  const int rbase = lo ? 0 : 8;
  const int wword = c0 >> 5;

#pragma unroll
  for (int v = 0; v < 8; ++v) {
    const int rv = rbase + v;
    const float rx1 = ROWS[0][rv], ry1 = ROWS[1][rv];
    const float rx2 = ROWS[2][rv], ry2 = ROWS[3][rv];
    const int rr = r0 + rv;
    // half 0 (cols c0..c0+15)
    float iw0 = fminf(ax2, rx2) - fmaxf(ax1, rx1) + 1.f;
    float ih0 = fminf(ay2, ry2) - fmaxf(ay1, ry1) + 1.f;
    float in0 = fmaxf(iw0, 0.f) * fmaxf(ih0, 0.f);
    bool s0 = (ca > rr) && ((1.f + NMS_T) * in0 > NMS_T * asum0[v]);
    unsigned bal0 = (unsigned)__ballot((int)s0);
    // half 1 (cols c0+16..c0+31)
    float iw1 = fminf(bx2, rx2) - fmaxf(bx1, rx1) + 1.f;
    float ih1 = fminf(by2, ry2) - fmaxf(by1, ry1) + 1.f;
    float in1 = fmaxf(iw1, 0.f) * fmaxf(ih1, 0.f);
    bool s1 = (cb > rr) && ((1.f + NMS_T) * in1 > NMS_T * asum1[v]);
    unsigned bal1 = (unsigned)__ballot((int)s1);
    // compose full 32-bit words: low lanes -> row r0+v, high lanes -> row r0+8+v
    if (lane == 0) {
      unsigned wlo = (bal0 & 0xFFFFu) | ((bal1 & 0xFFFFu) << 16);
      unsigned whi = (bal0 >> 16)     | (bal1 & 0xFFFF0000u);
      mask[(size_t)(r0 + v) * MASKW_ + wword] = wlo;
      mask[(size_t)(r0 + 8 + v) * MASKW_ + wword] = whi;
    }
  }
}

// ---------------- K5: sequential greedy reduction over bitmask ----------------
__global__ void __launch_bounds__(512)
nms_reduce(const unsigned* __restrict__ mask, const int* __restrict__ TVALID,
           int* __restrict__ keep_list, int* __restrict__ keep_count) {
  __shared__ unsigned remv[MASKW_];
  __shared__ int curKeep;
  __shared__ int cnt;
  for (int w = threadIdx.x; w < MASKW_; w += blockDim.x) remv[w] = 0u;
  if (threadIdx.x == 0) cnt = 0;
  __syncthreads();
  for (int i = 0; i < PRE_; ++i) {
    if (threadIdx.x == 0) {
      if (i + 1 < PRE_) __builtin_prefetch(mask + (size_t)(i + 1) * MASKW_, 0, 0);
      bool alive = TVALID[i] && !((remv[i >> 5] >> (i & 31)) & 1u);
      curKeep = alive ? 1 : 0;
      if (alive) { keep_list[cnt] = i; cnt++; }
    }
    __syncthreads();
    if (curKeep) {
      const unsigned* row = mask + (size_t)i * MASKW_;
      for (int w = threadIdx.x; w < MASKW_; w += blockDim.x) remv[w] |= row[w];
    }
    __syncthreads();
  }
  if (threadIdx.x == 0) *keep_count = cnt;
}

// ---------------- K6: emit (blob[2000x5], scores[2000x1]) ----------------
__global__ void write_out(const int* __restrict__ keep_list, const int* __restrict__ keep_count,
                          const float* __restrict__ TBOX, const float* __restrict__ TS,
                          float* __restrict__ out) {
  int t = blockIdx.x * blockDim.x + threadIdx.x;
  if (t >= POST_) return;
  int c = *keep_count;
  float x1 = 0.f, y1 = 0.f, x2 = 0.f, y2 = 0.f, s = 0.f;
  if (t < c) {
    int i = keep_list[t];
    x1 = TBOX[0 * PRE_ + i]; y1 = TBOX[1 * PRE_ + i];
    x2 = TBOX[2 * PRE_ + i]; y2 = TBOX[3 * PRE_ + i];
    s = TS[i];
  }
  out[t * 5 + 0] = 0.f;   // batch index
  out[t * 5 + 1] = x1;
  out[t * 5 + 2] = y1;
  out[t * 5 + 3] = x2;
  out[t * 5 + 4] = y2;
  out[POST_ * 5 + t] = s;
}

// ---------------- host: workspace carve + launch sequence ----------------
extern "C" void kernel_launch(void* const* d_in, const int* in_sizes, int n_in,
                              void* d_out, int out_size, void* d_ws, size_t ws_size,
                              hipStream_t stream) {
  const float* deltas = (const float*)d_in[0];   // (1,36,128,128)
  const float* scores = (const float*)d_in[1];   // (1,18,128,128)
  const float* iminfo = (const float*)d_in[2];   // (3,)
  float* out = (float*)d_out;                    // 2000*5 + 2000 floats

  char* ws = (char*)d_ws;
  size_t o = 0;
  auto carve = [&](size_t bytes) -> void* {
    void* p = ws + o;
    o = (o + bytes + 255) & ~(size_t)255;
    return p;
  };
  unsigned long long* keys = (unsigned long long*)carve((size_t)NPAD_ * 8);
  float* X1 = (float*)carve((size_t)N_ * 4);
  float* Y1 = (float*)carve((size_t)N_ * 4);
  float* X2 = (float*)carve((size_t)N_ * 4);
  float* Y2 = (float*)carve((size_t)N_ * 4);
  float* TBOX = (float*)carve((size_t)5 * PRE_ * 4);   // contiguous [5][12000]
  float* TS = (float*)carve((size_t)PRE_ * 4);
  int* TVALID = (int*)carve((size_t)PRE_ * 4);
  int* keep_list = (int*)carve((size_t)PRE_ * 4);
  int* keep_count = (int*)carve(256);
  unsigned* mask = (unsigned*)carve((size_t)PRE_ * MASKW_ * 4);   // 18 MB, fits L2

  // K0: zero mask + counter (diagonal-skipped tiles rely on zeros)
  {
    int n = PRE_ * MASKW_;
    zero_mask_kernel<<<(n + 255) / 256, 256, 0, stream>>>(mask, n, keep_count);
  }
  // K1: decode
  decode_kernel<<<NPAD_ / 256, 256, 0, stream>>>(deltas, scores, iminfo,
                                                 X1, Y1, X2, Y2, keys);
  // K2: bitonic sort (descending). LDS-resident for strides <= 1024:
  //     36 launches instead of 171; key array stays resident in L2.
  bitonic_local<<<NPAD_ / 2048, 1024, 0, stream>>>(keys, 0u);      // k = 2..2048
  for (unsigned k = 4096; k <= (unsigned)NPAD_; k <<= 1) {
    for (unsigned j = k >> 1; j >= 2048; j >>= 1)
      bitonic_step<<<NPAD_ / 256, 256, 0, stream>>>(keys, j, k);   // global strides
    bitonic_local<<<NPAD_ / 2048, 1024, 0, stream>>>(keys, k);     // j = 1024..1
  }
  // K3: gather top-12000
  gather_top<<<(PRE_ + 255) / 256, 256, 0, stream>>>(keys, X1, Y1, X2, Y2,
                                                     TBOX, TS, TVALID);
  // K4: IoU suppression bitmask: 750 row-tiles x 375 word-columns, 8 waves/block
  {
    dim3 grid((MASKW_ + 7) / 8, PRE_ / 16, 1);
    nms_mask_kernel<<<grid, 256, 0, stream>>>(TBOX, mask);
  }
  // K5: greedy reduction
  nms_reduce<<<1, 512, 0, stream>>>(mask, TVALID, keep_list, keep_count);
  // K6: final outputs
  write_out<<<(POST_ + 255) / 256, 256, 0, stream>>>(keep_list, keep_count,
                                                     TBOX, TS, out);
}